// net_supervised_loss_55533927137416
// MI455X (gfx1250) — compile-verified
//
#include <hip/hip_runtime.h>
#include <math.h>

#define NCLANS 600
#define MAXF   128
#define BSZ    8192
#define LSZ    64
#define DSZ    256
#define NEGV   (-1e9f)
#define XS     260   // padded LDS row stride (floats) -> conflict-free A gathers

typedef float v2f __attribute__((ext_vector_type(2)));
typedef float v4f __attribute__((ext_vector_type(4)));
typedef float v8f __attribute__((ext_vector_type(8)));

// ---------------- accumulator init ----------------
__global__ void k_init(float* acc) {
    if (threadIdx.x < 4) acc[threadIdx.x] = 0.0f;
}

// ---------------- mean over L: output[B,L,D] -> xbar[B,D] ----------------
// Streaming read of 537MB: non-temporal so it doesn't evict W_fam/xbar from L2.
__global__ __launch_bounds__(256) void k_mean(const float* __restrict__ out,
                                              float* __restrict__ xbar) {
    int i4 = blockIdx.x * 256 + threadIdx.x;       // float4 index into [B, D/4]
    int b  = i4 >> 6;                               // D/4 = 64
    int c4 = i4 & 63;
    const v4f* p = (const v4f*)out;
    long base = (long)b * LSZ * (DSZ / 4) + c4;
    v4f s = {0.f, 0.f, 0.f, 0.f};
    for (int l = 0; l < LSZ; ++l) {
        v4f v = __builtin_nontemporal_load(&p[base + (long)l * (DSZ / 4)]);
        s += v;
    }
    s *= (1.0f / (float)LSZ);
    ((v4f*)xbar)[i4] = s;
}

// ---------------- clan CE loss via f32 WMMA ----------------
// block = 16 rows of xbar, 128 threads = 4 waves; each wave sweeps 16-clan tiles.
__global__ __launch_bounds__(128) void k_clan(const float* __restrict__ xbar,
                                              const float* __restrict__ Wc,
                                              const float* __restrict__ bc,
                                              const int*   __restrict__ labc,
                                              float*       __restrict__ acc) {
    __shared__ float  sx[16 * XS];
    __shared__ float4 red[16 * 64];    // per-row: (max, sumexp, label-logit, pad)
    __shared__ int    slab[16];

    const int tid = threadIdx.x;
    const int b0  = blockIdx.x * 16;

    if (tid < 16) slab[tid] = labc[b0 + tid];

    // stage 16x256 x-tile into padded LDS (float4, coalesced)
    for (int i = 0; i < 8; ++i) {
        int idx = tid + i * 128;           // 0..1023 float4s over [16][64]
        int row = idx >> 6, k4 = idx & 63;
        float4 v = ((const float4*)xbar)[(long)(b0 + row) * 64 + k4];
        float* d = &sx[row * XS + k4 * 4];
        d[0] = v.x; d[1] = v.y; d[2] = v.z; d[3] = v.w;
    }
    __syncthreads();

    const int wave = tid >> 5;
    const int lane = tid & 31;
    const int ln   = lane & 15;
    const int half = lane >> 4;

    int labr[8];                            // row labels hoisted to registers
#pragma unroll
    for (int r = 0; r < 8; ++r) labr[r] = slab[r + 8 * half];

    float m[8], s[8], labv[8];
#pragma unroll
    for (int r = 0; r < 8; ++r) { m[r] = -INFINITY; s[r] = 0.0f; labv[r] = 0.0f; }

    for (int t = wave; t < 38; t += 4) {           // ceil(600/16)=38 tiles
        const int  col0  = t * 16;
        const int  n     = col0 + ln;
        const bool valid = (n < NCLANS);
        const int  nc    = valid ? n : (NCLANS - 1);   // clamp: branch-free loads

        v8f c = {0.f, 0.f, 0.f, 0.f, 0.f, 0.f, 0.f, 0.f};
        const float* wrow = Wc + (long)nc * DSZ;
        const float* xrow = &sx[ln * XS];
#pragma unroll 8
        for (int k0 = 0; k0 < DSZ; k0 += 4) {
            const int kk = k0 + 2 * half;
            v2f a;  a.x = xrow[kk]; a.y = xrow[kk + 1];
            v2f bb; bb.x = wrow[kk]; bb.y = wrow[kk + 1];
            // D = A(16x4 f32) x B(4x16 f32) + C, pure fp32 path
            c = __builtin_amdgcn_wmma_f32_16x16x4_f32(
                    false, a, false, bb, (short)0, c, false, false);
        }

        const float bias = bc[nc];
        if (valid) {
#pragma unroll
            for (int r = 0; r < 8; ++r) {
                float v = c[r] + bias;              // lane's column n, row r+8*half
                labv[r] += (n == labr[r]) ? v : 0.0f;   // branch-free label capture
                float nm = fmaxf(m[r], v);
                s[r] = s[r] * expf(m[r] - nm) + expf(v - nm);
                m[r] = nm;
            }
        }
    }

#pragma unroll
    for (int r = 0; r < 8; ++r) {
        int row = r + 8 * half;
        red[row * 64 + wave * 16 + ln] = make_float4(m[r], s[r], labv[r], 0.0f);
    }
    __syncthreads();

    if (tid < 16) {
        float M = -INFINITY, S = 0.0f, LV = 0.0f;
        for (int i = 0; i < 64; ++i) {
            float4 v = red[tid * 64 + i];
            LV += v.z;                              // exactly one contributor per row
            if (v.x != -INFINITY) {
                float nm = fmaxf(M, v.x);
                S = S * expf(M - nm) + v.y * expf(v.x - nm);
                M = nm;
            }
        }
        float nll = (M + logf(S)) - LV;
        atomicAdd(&acc[0], nll);
    }
}

// ---------------- routed family head loss (W_fam is L2-resident) ----------------
__global__ __launch_bounds__(128) void k_fam(const float* __restrict__ xbar,
                                             const float* __restrict__ Wf,
                                             const float* __restrict__ bf,
                                             const int*   __restrict__ labc,
                                             const int*   __restrict__ labf,
                                             const int*   __restrict__ fc,
                                             float*       __restrict__ acc) {
    const int b    = blockIdx.x;
    const int tid  = threadIdx.x;
    const int clan = labc[b];
    const int nf   = fc[clan];
    if (nf <= 1) return;                           // uniform across block

    __shared__ float sx[DSZ];
    __shared__ float lv[MAXF];

    if (tid < 64) ((float4*)sx)[tid] = ((const float4*)xbar)[(long)b * 64 + tid];
    __syncthreads();

    // thread f computes logit for family f
    const int f = tid;
    const float4* w4 = (const float4*)(Wf + ((long)clan * MAXF + f) * DSZ);
    const float4* x4 = (const float4*)sx;
    float a = 0.0f;
#pragma unroll 4
    for (int k = 0; k < 64; ++k) {
        float4 w = w4[k];
        float4 x = x4[k];                           // uniform address -> LDS broadcast
        a += w.x * x.x + w.y * x.y + w.z * x.z + w.w * x.w;
    }
    lv[f] = (f < nf) ? (a + bf[clan * MAXF + f]) : NEGV;
    __syncthreads();

    if (tid < 32) {                                 // wave32 logsumexp reduce
        float v0 = lv[tid], v1 = lv[tid + 32], v2 = lv[tid + 64], v3 = lv[tid + 96];
        float mx = fmaxf(fmaxf(v0, v1), fmaxf(v2, v3));
        for (int o = 16; o > 0; o >>= 1) mx = fmaxf(mx, __shfl_xor(mx, o, 32));
        float ss = expf(v0 - mx) + expf(v1 - mx) + expf(v2 - mx) + expf(v3 - mx);
        for (int o = 16; o > 0; o >>= 1) ss += __shfl_xor(ss, o, 32);
        if (tid == 0) {
            float nll = (mx + logf(ss)) - lv[labf[b]];
            atomicAdd(&acc[1], nll);
            atomicAdd(&acc[2], 1.0f);
        }
    }
}

// ---------------- final scalar ----------------
__global__ void k_final(const float* __restrict__ acc, float* __restrict__ out) {
    if (threadIdx.x == 0)
        out[0] = (acc[0] + acc[1]) / ((float)BSZ + acc[2]);
}

extern "C" void kernel_launch(void* const* d_in, const int* in_sizes, int n_in,
                              void* d_out, int out_size, void* d_ws, size_t ws_size,
                              hipStream_t stream) {
    (void)in_sizes; (void)n_in; (void)out_size; (void)ws_size;
    const float* output = (const float*)d_in[0];
    const int*   labc   = (const int*)  d_in[1];
    const int*   labf   = (const int*)  d_in[2];
    const int*   fc     = (const int*)  d_in[3];
    const float* Wc     = (const float*)d_in[4];
    const float* bc     = (const float*)d_in[5];
    const float* Wf     = (const float*)d_in[6];
    const float* bf     = (const float*)d_in[7];

    float* acc  = (float*)d_ws;          // [0]=loss_clan [1]=loss_family [2]=count
    float* xbar = (float*)d_ws + 4;      // [B, D] fp32, 16B aligned

    k_init <<<1, 4, 0, stream>>>(acc);
    k_mean <<<(BSZ * DSZ / 4) / 256, 256, 0, stream>>>(output, xbar);
    k_clan <<<BSZ / 16, 128, 0, stream>>>(xbar, Wc, bc, labc, acc);
    k_fam  <<<BSZ, 128, 0, stream>>>(xbar, Wf, bf, labc, labf, fc, acc);
    k_final<<<1, 1, 0, stream>>>(acc, (float*)d_out);
}